// RelGraphConvLayer_1520418423097
// MI455X (gfx1250) — compile-verified
//
#include <hip/hip_runtime.h>
#include <math.h>

#define DIN 128
#define DOUT 128
#define NEG_SLOPE 0.2f

typedef __attribute__((ext_vector_type(2))) float v2f;
typedef __attribute__((ext_vector_type(8))) float v8f;

// ---- monotone float<->uint encoding so atomicMax(uint) orders floats (incl. negatives) ----
__device__ __forceinline__ unsigned f2mono(float f) {
  unsigned u = __float_as_uint(f);
  return (u & 0x80000000u) ? ~u : (u | 0x80000000u);
}
__device__ __forceinline__ float mono2f(unsigned u) {
  return (u & 0x80000000u) ? __uint_as_float(u & 0x7FFFFFFFu) : __uint_as_float(~u);
}

// ---------------- init: zero output, segment max (encoded), segment sum ----------------
__global__ void init_bufs(float* __restrict__ out, unsigned* __restrict__ smax,
                          float* __restrict__ ssum, int outN, int RN) {
  int idx = blockIdx.x * blockDim.x + threadIdx.x;
  if (idx < outN) out[idx] = 0.0f;
  if (idx < RN) { smax[idx] = 0u; ssum[idx] = 0.0f; }
}

// ---------------- fold attention vectors: vl[r,d] = sum_e Wsrc[r,d,e]*attn_l[r,e] ----------------
__global__ void make_attn_vec(const float* __restrict__ Ws, const float* __restrict__ Wd,
                              const float* __restrict__ al, const float* __restrict__ ar,
                              float* __restrict__ vl, float* __restrict__ vr, int total) {
  int idx = blockIdx.x * blockDim.x + threadIdx.x;
  if (idx >= total) return;
  int r = idx / DIN, d = idx % DIN;
  const float* ws = Ws + ((size_t)r * DIN + d) * DOUT;
  const float* wd = Wd + ((size_t)r * DIN + d) * DOUT;
  const float* a_l = al + r * DOUT;
  const float* a_r = ar + r * DOUT;
  float sl = 0.f, sr = 0.f;
  for (int e = 0; e < DOUT; ++e) { sl += ws[e] * a_l[e]; sr += wd[e] * a_r[e]; }
  vl[idx] = sl;
  vr[idx] = sr;
}

// ---------------- hs[r] = x @ W_src[r] via V_WMMA_F32_16X16X4_F32 ----------------
// One wave computes a full 16x128 row strip (8 col tiles) of hs for relation blockIdx.y.
// A fragment (16x4 f32): lanes 0-15 -> M=lane, VGPR0=K0,VGPR1=K1; lanes 16-31 -> K2,K3.
// B fragment (4x16 f32): lanes 0-15 -> N=lane, VGPR0=K0,VGPR1=K1; lanes 16-31 -> K2,K3.
// C/D (16x16 f32): lanes 0-15 rows 0-7 in v0..7; lanes 16-31 rows 8-15.
__global__ void proj_gemm_wmma(const float* __restrict__ x,
                               const float* __restrict__ W,   // [R, DIN, DOUT]
                               float* __restrict__ hs,        // [R, N, DOUT]
                               int N, int rowTiles) {
  const int wave = threadIdx.x >> 5;
  const int lane = threadIdx.x & 31;
  const int rowTile = blockIdx.x * (blockDim.x >> 5) + wave;
  if (rowTile >= rowTiles) return;      // wave-uniform exit: EXEC all-ones at WMMA
  const int r = blockIdx.y;
  const int half = lane >> 4;           // 0: K pair {0,1}; 1: K pair {2,3}
  const int lid  = lane & 15;
  const int row0 = rowTile * 16;

  int arow = row0 + lid;
  if (arow >= N) arow = N - 1;          // clamp loads for ragged last tile
  const float* __restrict__ xa = x + (size_t)arow * DIN + 2 * half;
  const float* __restrict__ Wr = W + (size_t)r * DIN * DOUT;

  v8f acc[8] = {};
  for (int k = 0; k < DIN; k += 4) {
    v2f a = *(const v2f*)(xa + k);      // 8B-aligned: (2*half + k) even
    const float* __restrict__ Wk = Wr + (size_t)(k + 2 * half) * DOUT + lid;
#pragma unroll
    for (int t = 0; t < 8; ++t) {
      v2f b;
      b.x = Wk[t * 16];                 // K = k+2*half,   col = t*16+lid
      b.y = Wk[t * 16 + DOUT];          // K = k+2*half+1
      acc[t] = __builtin_amdgcn_wmma_f32_16x16x4_f32(
          false, a, false, b, (short)0, acc[t], false, false);
    }
  }

  float* __restrict__ hrow = hs + (size_t)r * N * DOUT;
#pragma unroll
  for (int t = 0; t < 8; ++t) {
#pragma unroll
    for (int v = 0; v < 8; ++v) {
      int row = row0 + half * 8 + v;
      if (row < N) hrow[(size_t)row * DOUT + t * 16 + lid] = acc[t][v];
    }
  }
}

// ---------------- el[r,n] = x[n].vl[r], er[r,n] = x[n].vr[r]; one wave per node ----------------
__global__ void attn_feat(const float* __restrict__ x,
                          const float* __restrict__ vl, const float* __restrict__ vr,
                          float* __restrict__ el, float* __restrict__ er, int N, int R) {
  const int wave = threadIdx.x >> 5;
  const int lane = threadIdx.x & 31;
  const int node = blockIdx.x * (blockDim.x >> 5) + wave;
  if (node >= N) return;
  const float4 xv = *(const float4*)(x + (size_t)node * DIN + lane * 4);
  for (int r = 0; r < R; ++r) {
    const float4 lv = *(const float4*)(vl + r * DIN + lane * 4);
    const float4 rv = *(const float4*)(vr + r * DIN + lane * 4);
    float pl = xv.x * lv.x + xv.y * lv.y + xv.z * lv.z + xv.w * lv.w;
    float pr = xv.x * rv.x + xv.y * rv.y + xv.z * rv.z + xv.w * rv.w;
#pragma unroll
    for (int m = 16; m >= 1; m >>= 1) {
      pl += __shfl_xor(pl, m, 32);
      pr += __shfl_xor(pr, m, 32);
    }
    if (lane == 0) {
      el[(size_t)r * N + node] = pl;
      er[(size_t)r * N + node] = pr;
    }
  }
}

// ---------------- edge pass 1: leaky-relu logits + segment max ----------------
__global__ void edge_logits_max(const int* __restrict__ src, const int* __restrict__ dst,
                                const float* __restrict__ el, const float* __restrict__ er,
                                float* __restrict__ lbuf, unsigned* __restrict__ smax,
                                int N, int E, int RE) {
  int idx = blockIdx.x * blockDim.x + threadIdx.x;
  if (idx >= RE) return;
  int rid = idx / E;
  int s = src[idx], d = dst[idx];
  float l = el[(size_t)rid * N + s] + er[(size_t)rid * N + d];
  l = (l > 0.f) ? l : NEG_SLOPE * l;
  lbuf[idx] = l;
  atomicMax(smax + (size_t)rid * N + d, f2mono(l));
}

// ---------------- edge pass 2: a = exp(l - max) + segment sum ----------------
__global__ void edge_expsum(const int* __restrict__ dst,
                            float* __restrict__ abuf, const unsigned* __restrict__ smax,
                            float* __restrict__ ssum, int N, int E, int RE) {
  int idx = blockIdx.x * blockDim.x + threadIdx.x;
  if (idx >= RE) return;
  int rid = idx / E;
  int d = dst[idx];
  float m = mono2f(smax[(size_t)rid * N + d]);
  float a = expf(abuf[idx] - m);
  abuf[idx] = a;
  atomicAdd(ssum + (size_t)rid * N + d, a);
}

// ---------------- edge pass 3: out[dst] += alpha * hs[r, src]; one wave per edge ----------------
__global__ void edge_scatter(const int* __restrict__ src, const int* __restrict__ dst,
                             const float* __restrict__ abuf, const float* __restrict__ ssum,
                             const float* __restrict__ hs, float* __restrict__ out,
                             int N, int E, int RE) {
  const int wave = threadIdx.x >> 5;
  const int lane = threadIdx.x & 31;
  const int idx = blockIdx.x * (blockDim.x >> 5) + wave;
  if (idx >= RE) return;
  const int rid = idx / E;
  const int s = src[idx], d = dst[idx];
  const float alpha = abuf[idx] / ssum[(size_t)rid * N + d];
  const float* __restrict__ hrow = hs + ((size_t)rid * N + s) * DOUT;
  float* __restrict__ orow = out + (size_t)d * DOUT;
#pragma unroll
  for (int j = 0; j < DOUT / 32; ++j)
    atomicAdd(orow + j * 32 + lane, alpha * hrow[j * 32 + lane]);
}

// ---------------- epilogue: relu(out + sum_r gat_bias[r] + h_bias) ----------------
__global__ void bias_relu(float* __restrict__ out, const float* __restrict__ gb,
                          const float* __restrict__ hb, int total, int R) {
  int idx = blockIdx.x * blockDim.x + threadIdx.x;
  if (idx >= total) return;
  int d = idx % DOUT;
  float b = hb[d];
  for (int r = 0; r < R; ++r) b += gb[r * DOUT + d];
  float v = out[idx] + b;
  out[idx] = (v > 0.f) ? v : 0.f;
}

extern "C" void kernel_launch(void* const* d_in, const int* in_sizes, int n_in,
                              void* d_out, int out_size, void* d_ws, size_t ws_size,
                              hipStream_t stream) {
  const float* x      = (const float*)d_in[0];
  const int*   esrc   = (const int*)d_in[1];
  const int*   edst   = (const int*)d_in[2];
  const float* Wsrc   = (const float*)d_in[3];
  const float* Wdst   = (const float*)d_in[4];
  const float* attn_l = (const float*)d_in[5];
  const float* attn_r = (const float*)d_in[6];
  const float* gbias  = (const float*)d_in[7];
  const float* hbias  = (const float*)d_in[8];
  float* out = (float*)d_out;

  const int N  = in_sizes[0] / DIN;
  const int R  = in_sizes[3] / (DIN * DOUT);
  const int E  = in_sizes[1] / R;
  const int RN = R * N;
  const int RE = R * E;

  // workspace carve (floats)
  float* ws = (float*)d_ws;
  size_t off = 0;
  float*    hs   = ws + off; off += (size_t)R * N * DOUT;  // 204.8 MB
  float*    el   = ws + off; off += (size_t)RN;
  float*    er   = ws + off; off += (size_t)RN;
  unsigned* smax = (unsigned*)(ws + off); off += (size_t)RN;
  float*    ssum = ws + off; off += (size_t)RN;
  float*    abuf = ws + off; off += (size_t)RE;
  float*    vl   = ws + off; off += (size_t)R * DIN;
  float*    vr   = ws + off; off += (size_t)R * DIN;

  const int initTotal = (N * DOUT > RN) ? N * DOUT : RN;
  init_bufs<<<(initTotal + 255) / 256, 256, 0, stream>>>(out, smax, ssum, N * DOUT, RN);

  make_attn_vec<<<(R * DIN + 255) / 256, 256, 0, stream>>>(Wsrc, Wdst, attn_l, attn_r, vl, vr, R * DIN);

  const int rowTiles = (N + 15) / 16;
  dim3 ggrid((rowTiles + 7) / 8, R);
  proj_gemm_wmma<<<ggrid, 256, 0, stream>>>(x, Wsrc, hs, N, rowTiles);

  attn_feat<<<(N + 7) / 8, 256, 0, stream>>>(x, vl, vr, el, er, N, R);

  edge_logits_max<<<(RE + 255) / 256, 256, 0, stream>>>(esrc, edst, el, er, abuf, smax, N, E, RE);
  edge_expsum<<<(RE + 255) / 256, 256, 0, stream>>>(edst, abuf, smax, ssum, N, E, RE);
  edge_scatter<<<(RE + 7) / 8, 256, 0, stream>>>(esrc, edst, abuf, ssum, hs, out, N, E, RE);

  bias_relu<<<(N * DOUT + 255) / 256, 256, 0, stream>>>(out, gbias, hbias, N * DOUT, R);
}